// Model_25056839204944
// MI455X (gfx1250) — compile-verified
//
#include <hip/hip_runtime.h>
#include <hip/hip_bf16.h>
#include <math.h>

// ---------------------------------------------------------------------------
// Problem constants (from reference)
// ---------------------------------------------------------------------------
#define NS      4      // NUM_STREAMS
#define DIM     256
#define TILE_M  16     // batch rows per workgroup (one WMMA M tile)
#define SK_ITERS 10
#define SK_TAU   0.05f

typedef __attribute__((ext_vector_type(2))) float v2f;
typedef __attribute__((ext_vector_type(8))) float v8f;

// ---------------------------------------------------------------------------
// Kernel 1: tiny scalar kernel — sinkhorn on 4x4 logits + two softmaxes.
// Writes 24 floats to workspace:
//   coef[0..15]  = h_res[s][t]   (row-major, s*4+t)
//   coef[16..19] = h_pre[s]
//   coef[20..23] = h_post[t]
// ---------------------------------------------------------------------------
__global__ void sinkhorn_coef_kernel(const float* __restrict__ h_res_logits,
                                     const float* __restrict__ h_pre_logits,
                                     const float* __restrict__ h_post_logits,
                                     float* __restrict__ coef) {
    if (threadIdx.x != 0) return;

    float z[NS][NS];
    #pragma unroll
    for (int i = 0; i < NS; ++i)
        #pragma unroll
        for (int j = 0; j < NS; ++j)
            z[i][j] = h_res_logits[i * NS + j] / SK_TAU;

    float u[NS] = {0.f, 0.f, 0.f, 0.f};
    float v[NS] = {0.f, 0.f, 0.f, 0.f};
    const float log_marginal = -logf((float)NS);

    for (int it = 0; it < SK_ITERS; ++it) {
        // u = logm - logsumexp_j(z[i][j] + v[j])
        for (int i = 0; i < NS; ++i) {
            float mx = -1e30f;
            for (int j = 0; j < NS; ++j) { float t = z[i][j] + v[j]; mx = fmaxf(mx, t); }
            float s = 0.f;
            for (int j = 0; j < NS; ++j) s += expf(z[i][j] + v[j] - mx);
            u[i] = log_marginal - (mx + logf(s));
        }
        // v = logm - logsumexp_i(z[i][j] + u[i])
        for (int j = 0; j < NS; ++j) {
            float mx = -1e30f;
            for (int i = 0; i < NS; ++i) { float t = z[i][j] + u[i]; mx = fmaxf(mx, t); }
            float s = 0.f;
            for (int i = 0; i < NS; ++i) s += expf(z[i][j] + u[i] - mx);
            v[j] = log_marginal - (mx + logf(s));
        }
    }

    #pragma unroll
    for (int i = 0; i < NS; ++i)
        #pragma unroll
        for (int j = 0; j < NS; ++j)
            coef[i * NS + j] = expf(z[i][j] + u[i] + v[j]) * (float)NS;

    // h_pre = softmax(h_pre_logits)
    {
        float mx = -1e30f;
        for (int i = 0; i < NS; ++i) mx = fmaxf(mx, h_pre_logits[i]);
        float s = 0.f, e[NS];
        for (int i = 0; i < NS; ++i) { e[i] = expf(h_pre_logits[i] - mx); s += e[i]; }
        for (int i = 0; i < NS; ++i) coef[16 + i] = e[i] / s;
    }
    // h_post = softmax(h_post_logits)
    {
        float mx = -1e30f;
        for (int i = 0; i < NS; ++i) mx = fmaxf(mx, h_post_logits[i]);
        float s = 0.f, e[NS];
        for (int i = 0; i < NS; ++i) { e[i] = expf(h_post_logits[i] - mx); s += e[i]; }
        for (int i = 0; i < NS; ++i) coef[20 + i] = e[i] / s;
    }
}

// ---------------------------------------------------------------------------
// Kernel 2: fused per-16-row tile.
//  Phase A: s_bi[m][k] = sum_s h_pre[s] * res[b0+m, s, k]           (LDS, 16KB)
//  Phase B: y = s_bi @ W^T via V_WMMA_F32_16X16X4_F32, y -> s_y     (LDS, 16KB)
//  Phase C: out[b,t,d] = sum_s h_res[s][t]*res[b,s,d] + h_post[t]*y[b,d]
// 128 threads = 4 waves; wave w owns output columns [w*64, w*64+64).
// ---------------------------------------------------------------------------
__global__ __launch_bounds__(128) void fused_mix_gemm_kernel(
    const float* __restrict__ residuals,   // (B, NS, DIM)
    const float* __restrict__ W,           // (DIM, DIM) row-major: W[n][k]
    const float* __restrict__ coef,        // 24 floats (see above)
    float* __restrict__ out) {             // (B, NS, DIM)

    __shared__ float s_bi[TILE_M][DIM];    // branch_input tile
    __shared__ float s_y [TILE_M][DIM];    // branch_output tile

    const int tid  = threadIdx.x;
    const int lane = tid & 31;
    const int wave = tid >> 5;
    const int b0   = blockIdx.x * TILE_M;

    // ---- Phase A: branch_input tile into LDS ----
    float hpre[NS];
    #pragma unroll
    for (int s = 0; s < NS; ++s) hpre[s] = coef[16 + s];

    // 16 rows * 64 float4-chunks = 1024 chunks, 8 per thread
    #pragma unroll
    for (int c = tid; c < TILE_M * (DIM / 4); c += 128) {
        const int m  = c >> 6;              // / 64
        const int d4 = (c & 63) << 2;       // * 4
        const float* base = residuals + ((size_t)(b0 + m) * NS) * DIM + d4;
        float ax = 0.f, ay = 0.f, az = 0.f, aw = 0.f;
        #pragma unroll
        for (int s = 0; s < NS; ++s) {
            const float4 r = *(const float4*)(base + (size_t)s * DIM);
            ax += hpre[s] * r.x;
            ay += hpre[s] * r.y;
            az += hpre[s] * r.z;
            aw += hpre[s] * r.w;
        }
        float4* dst = (float4*)&s_bi[m][d4];
        *dst = make_float4(ax, ay, az, aw);
    }
    __syncthreads();

    // ---- Phase B: GEMM y[m][n] = sum_k bi[m][k] * W[n][k] ----
    // A-frag layout (16x4 f32): lanes 0-15 -> M=lane, VGPR0/1 = K,K+1
    //                           lanes 16-31 -> M=lane-16, VGPR0/1 = K+2,K+3
    // B-frag layout (4x16 f32): lane -> N, VGPRs hold K pair (mirror of A)
    const int mrow  = lane & 15;
    const int khalf = (lane >> 4) << 1;     // 0 or 2

    v8f acc[4];
    #pragma unroll
    for (int t = 0; t < 4; ++t) acc[t] = (v8f)(0.f);

    for (int k0 = 0; k0 < DIM; k0 += 4) {
        const int kb = k0 + khalf;
        const v2f afrag = *(const v2f*)&s_bi[mrow][kb];           // ds_load_b64
        #pragma unroll
        for (int t = 0; t < 4; ++t) {
            const int n0 = (wave * 4 + t) * 16;
            const v2f bfrag = *(const v2f*)(W + (size_t)(n0 + mrow) * DIM + kb);
            acc[t] = __builtin_amdgcn_wmma_f32_16x16x4_f32(
                false, afrag, false, bfrag, (short)0, acc[t], false, false);
        }
    }

    // Spill C/D accumulators to LDS: VGPR r of lane -> M = r + 8*(lane>>4), N = lane&15
    #pragma unroll
    for (int t = 0; t < 4; ++t) {
        const int n = (wave * 4 + t) * 16 + mrow;
        #pragma unroll
        for (int r = 0; r < 8; ++r) {
            s_y[r + ((lane >> 4) << 3)][n] = acc[t][r];
        }
    }
    __syncthreads();

    // ---- Phase C: mixing + depth-out, fused write ----
    float hres[NS][NS];
    #pragma unroll
    for (int s = 0; s < NS; ++s)
        #pragma unroll
        for (int t = 0; t < NS; ++t)
            hres[s][t] = coef[s * NS + t];
    float hpost[NS];
    #pragma unroll
    for (int t = 0; t < NS; ++t) hpost[t] = coef[20 + t];

    #pragma unroll
    for (int c = tid; c < TILE_M * (DIM / 4); c += 128) {
        const int m  = c >> 6;
        const int d4 = (c & 63) << 2;
        const float* base = residuals + ((size_t)(b0 + m) * NS) * DIM + d4;
        float4 r[NS];
        #pragma unroll
        for (int s = 0; s < NS; ++s) r[s] = *(const float4*)(base + (size_t)s * DIM);
        const float4 y = *(const float4*)&s_y[m][d4];

        #pragma unroll
        for (int t = 0; t < NS; ++t) {
            float ox = hpost[t] * y.x;
            float oy = hpost[t] * y.y;
            float oz = hpost[t] * y.z;
            float ow = hpost[t] * y.w;
            #pragma unroll
            for (int s = 0; s < NS; ++s) {
                ox += hres[s][t] * r[s].x;
                oy += hres[s][t] * r[s].y;
                oz += hres[s][t] * r[s].z;
                ow += hres[s][t] * r[s].w;
            }
            float4* dst = (float4*)(out + ((size_t)(b0 + m) * NS + t) * DIM + d4);
            *dst = make_float4(ox, oy, oz, ow);
        }
    }
}

// ---------------------------------------------------------------------------
// Launch
// ---------------------------------------------------------------------------
extern "C" void kernel_launch(void* const* d_in, const int* in_sizes, int n_in,
                              void* d_out, int out_size, void* d_ws, size_t ws_size,
                              hipStream_t stream) {
    const float* residuals     = (const float*)d_in[0];   // (B, NS, DIM)
    const float* h_res_logits  = (const float*)d_in[1];   // (NS, NS)
    const float* h_pre_logits  = (const float*)d_in[2];   // (NS,)
    const float* h_post_logits = (const float*)d_in[3];   // (NS,)
    const float* W_branch      = (const float*)d_in[4];   // (DIM, DIM)
    float*       out           = (float*)d_out;
    float*       coef          = (float*)d_ws;            // 24 floats

    const int B = in_sizes[0] / (NS * DIM);               // 65536

    sinkhorn_coef_kernel<<<1, 32, 0, stream>>>(h_res_logits, h_pre_logits,
                                               h_post_logits, coef);

    const int grid = B / TILE_M;                          // B is a multiple of 16
    fused_mix_gemm_kernel<<<grid, 128, 0, stream>>>(residuals, W_branch, coef, out);
}